// IDGIN_73882027425872
// MI455X (gfx1250) — compile-verified
//
#include <hip/hip_runtime.h>
#include <stdint.h>

typedef __attribute__((ext_vector_type(2))) float v2f;
typedef __attribute__((ext_vector_type(8))) float v8f;

#define DIM 128
#define HID 512
#define MT  2           // M-tiles (of 16 rows) per block -> 32 rows/block
#define HS_STRIDE 132   // 128 + 4 pad words  (stride % 64 = 4 -> conflict-free)
#define ZS_STRIDE 516   // 512 + 4 pad words

// Use CDNA5 async global->LDS DMA for staging (device pass only).
#if defined(__gfx1250__)
#define USE_ASYNC_LDS 1
#endif

// ---------------------------------------------------------------------------
// Kernel 1: h = x  (vectorized copy; merges the (1+eps)*x term, eps==0)
// ---------------------------------------------------------------------------
__global__ void idgin_copy_kernel(const float4* __restrict__ x,
                                  float4* __restrict__ h, int n4) {
    int i = blockIdx.x * blockDim.x + threadIdx.x;
    if (i < n4) h[i] = x[i];
}

// ---------------------------------------------------------------------------
// Kernel 2: h[row] += x[col] per edge. One wave (32 lanes) per edge; each lane
// handles a contiguous float4 chunk -> coalesced 512B reads + contiguous
// global_atomic_add_f32. x and h are L2-resident (51MB each vs 192MB L2).
// ---------------------------------------------------------------------------
__global__ void idgin_edge_kernel(const float* __restrict__ x,
                                  const int* __restrict__ rows,
                                  const int* __restrict__ cols,
                                  float* __restrict__ h, int nedges) {
    int e = blockIdx.x * 8 + (threadIdx.x >> 5);
    if (e >= nedges) return;
    int lane = threadIdx.x & 31;
    int r = rows[e];
    int c = cols[e];
    float4 v = ((const float4*)(x + (size_t)c * DIM))[lane];
    float* dst = h + (size_t)r * DIM + lane * 4;
    atomicAdd(dst + 0, v.x);
    atomicAdd(dst + 1, v.y);
    atomicAdd(dst + 2, v.z);
    atomicAdd(dst + 3, v.w);
}

// ---------------------------------------------------------------------------
// Kernel 3/4: fused 2-layer MLP over 32-row blocks using f32 WMMA 16x16x4.
// 8 waves/block. Each wave owns 2 M-tiles so every B-fragment feeds 2 WMMAs.
//   layer1: z = relu(h_tile @ W1 + b1), wave w covers cols [64w,64w+64)
//   layer2: out = z @ W2 + b2,          wave w covers cols [16w,16w+16)
// idx==nullptr: direct rows + plain store; idx!=nullptr: gathered rows +
// atomicAdd scatter (id_index duplicates must accumulate).
// ---------------------------------------------------------------------------
__global__ __launch_bounds__(256)
void idgin_mlp_kernel(const float* __restrict__ h,
                      const float* __restrict__ W1, const float* __restrict__ b1,
                      const float* __restrict__ W2, const float* __restrict__ b2,
                      const int* __restrict__ idx,
                      float* __restrict__ out, int nrows) {
    __shared__ float hs[MT * 16 * HS_STRIDE];   // 16.9 KB
    __shared__ float zs[MT * 16 * ZS_STRIDE];   // 66.0 KB

    const int tid  = threadIdx.x;
    const int w    = tid >> 5;        // wave id 0..7
    const int lane = tid & 31;
    const int m16  = lane & 15;
    const int hi   = lane >> 4;       // K-pair select / M-half select
    const int base = blockIdx.x * (MT * 16);

    // ---- stage 32 rows of h into LDS (async DMA global->LDS) ----
    for (int i = tid; i < MT * 16 * 32; i += 256) {
        int row = i >> 5;
        int c4  = i & 31;
        int sr  = base + row;
        sr = sr < nrows ? sr : nrows - 1;
        if (idx) sr = idx[sr];
        const float* gp = h + (size_t)sr * DIM + c4 * 4;
        float*       lp = &hs[row * HS_STRIDE + c4 * 4];
#if USE_ASYNC_LDS
        // 16B DMA straight into LDS; low 32 bits of flat LDS pointer are the
        // offset within the workgroup's LDS allocation (ISA aperture rule).
        asm volatile("global_load_async_to_lds_b128 %0, %1, off"
                     :: "v"((uint32_t)(uintptr_t)lp),
                        "v"((uint64_t)(uintptr_t)gp)
                     : "memory");
#else
        *(float4*)lp = *(const float4*)gp;
#endif
    }
#if USE_ASYNC_LDS
    asm volatile("s_wait_asynccnt 0x0" ::: "memory");
#endif
    __syncthreads();

    // ---- layer 1: K = 128 (32 WMMA k-steps), 2 M-tiles x 4 N-tiles ----
    v8f acc[MT][4] = {};
    for (int kk = 0; kk < 32; ++kk) {
        const int k0 = kk * 4 + hi * 2;
        v2f a[MT];
#pragma unroll
        for (int mt = 0; mt < MT; ++mt) {
            a[mt].x = hs[(mt * 16 + m16) * HS_STRIDE + k0];
            a[mt].y = hs[(mt * 16 + m16) * HS_STRIDE + k0 + 1];
        }
#pragma unroll
        for (int t = 0; t < 4; ++t) {
            const int colb = w * 64 + t * 16 + m16;
            v2f b;
            b.x = W1[(size_t)k0 * HID + colb];
            b.y = W1[(size_t)(k0 + 1) * HID + colb];
#pragma unroll
            for (int mt = 0; mt < MT; ++mt)
                acc[mt][t] = __builtin_amdgcn_wmma_f32_16x16x4_f32(
                    false, a[mt], false, b, (short)0, acc[mt][t], false, false);
        }
    }
    // epilogue: bias + ReLU -> zs
#pragma unroll
    for (int t = 0; t < 4; ++t) {
        const int colb = w * 64 + t * 16 + m16;
        const float bv = b1[colb];
#pragma unroll
        for (int mt = 0; mt < MT; ++mt)
#pragma unroll
            for (int r = 0; r < 8; ++r) {
                float v = acc[mt][t][r] + bv;
                zs[(mt * 16 + hi * 8 + r) * ZS_STRIDE + colb] = v > 0.f ? v : 0.f;
            }
    }
    __syncthreads();

    // ---- layer 2: K = 512 (128 WMMA k-steps), 2 M-tiles x 1 N-tile ----
    v8f acc2[MT] = {};
    const int colb2 = w * 16 + m16;
    for (int kk = 0; kk < 128; ++kk) {
        const int k0 = kk * 4 + hi * 2;
        v2f b;
        b.x = W2[(size_t)k0 * DIM + colb2];
        b.y = W2[(size_t)(k0 + 1) * DIM + colb2];
#pragma unroll
        for (int mt = 0; mt < MT; ++mt) {
            v2f a;
            a.x = zs[(mt * 16 + m16) * ZS_STRIDE + k0];
            a.y = zs[(mt * 16 + m16) * ZS_STRIDE + k0 + 1];
            acc2[mt] = __builtin_amdgcn_wmma_f32_16x16x4_f32(
                false, a, false, b, (short)0, acc2[mt], false, false);
        }
    }
    const float b2v = b2[colb2];
#pragma unroll
    for (int mt = 0; mt < MT; ++mt)
#pragma unroll
        for (int r = 0; r < 8; ++r) {
            const int row = base + mt * 16 + hi * 8 + r;
            if (row < nrows) {
                float v = acc2[mt][r] + b2v;
                if (idx) {
                    atomicAdd(&out[(size_t)idx[row] * DIM + colb2], v);
                } else {
                    out[(size_t)row * DIM + colb2] = v;
                }
            }
        }
}

// ---------------------------------------------------------------------------
extern "C" void kernel_launch(void* const* d_in, const int* in_sizes, int n_in,
                              void* d_out, int out_size, void* d_ws, size_t ws_size,
                              hipStream_t stream) {
    const float* x    = (const float*)d_in[0];
    const float* W1   = (const float*)d_in[1];
    const float* b1   = (const float*)d_in[2];
    const float* W2   = (const float*)d_in[3];
    const float* b2   = (const float*)d_in[4];
    const float* Wid1 = (const float*)d_in[5];
    const float* bid1 = (const float*)d_in[6];
    const float* Wid2 = (const float*)d_in[7];
    const float* bid2 = (const float*)d_in[8];
    const int*   eix  = (const int*)d_in[9];   // [2, E] flattened
    const int*   iix  = (const int*)d_in[10];  // [NID]

    const int Nn  = in_sizes[0] / DIM;
    const int Ee  = in_sizes[9] / 2;
    const int Nid = in_sizes[10];

    float* h = (float*)d_ws;   // [Nn, 128] f32 scratch (51.2 MB)

    // 1) h = x
    const int n4 = Nn * (DIM / 4);
    idgin_copy_kernel<<<(n4 + 255) / 256, 256, 0, stream>>>(
        (const float4*)x, (float4*)h, n4);

    // 2) h[row] += x[col] over all edges (one wave per edge)
    idgin_edge_kernel<<<(Ee + 7) / 8, 256, 0, stream>>>(
        x, eix, eix + Ee, h, Ee);

    // 3) out = MLP(h)  (dense, all N rows)
    idgin_mlp_kernel<<<(Nn + MT * 16 - 1) / (MT * 16), 256, 0, stream>>>(
        h, W1, b1, W2, b2, nullptr, (float*)d_out, Nn);

    // 4) out[id_index] += MLP_id(h[id_index])  (gather -> MLP -> atomic scatter)
    idgin_mlp_kernel<<<(Nid + MT * 16 - 1) / (MT * 16), 256, 0, stream>>>(
        h, Wid1, bid1, Wid2, bid2, iix, (float*)d_out, Nid);
}